// AIGerConv_50775103373989
// MI455X (gfx1250) — compile-verified
//
#include <hip/hip_runtime.h>

#define NNODES 100000
#define DIM    128
#define NREL   4
#define NEDGES 500000

typedef __attribute__((ext_vector_type(2))) float v2f;
typedef __attribute__((ext_vector_type(8))) float v8f;

// ---------------------------------------------------------------------------
// Phase 1: scatter raw features into per-relation accumulators.
// One wave32 per (relation, edge). Lane l handles 4 consecutive floats, so a
// wave reads a coalesced 512B row x[src] and issues 4 f32 atomics per lane
// into h[r][tgt]. x (51MB) and h (205MB) are ~resident in the 192MB L2, so
// the 256M atomics stay on-chip.
// ---------------------------------------------------------------------------
__global__ void rgcn_scatter_kernel(const float* __restrict__ x,
                                    const int* __restrict__ edges, // [NREL][2][NEDGES]
                                    float* __restrict__ h,         // [NREL][NNODES][DIM]
                                    float* __restrict__ deg) {     // [NREL][NNODES]
  const int r    = blockIdx.y;
  const int gid  = blockIdx.x * blockDim.x + threadIdx.x;
  const int e    = gid >> 5;          // exact launch: e < NEDGES always
  const int lane = gid & 31;

  const int src = edges[(size_t)r * 2 * NEDGES + e];
  const int tgt = edges[(size_t)r * 2 * NEDGES + NEDGES + e];

  const float4 v = *reinterpret_cast<const float4*>(x + (size_t)src * DIM + lane * 4);
  float* dst = h + ((size_t)r * NNODES + tgt) * DIM + lane * 4;
  atomicAdd(dst + 0, v.x);
  atomicAdd(dst + 1, v.y);
  atomicAdd(dst + 2, v.z);
  atomicAdd(dst + 3, v.w);
  if (lane == 0) atomicAdd(deg + (size_t)r * NNODES + tgt, 1.0f);
}

// ---------------------------------------------------------------------------
// Phase 2: out = Σ_r h_r·W_rᵀ + deg_r⊙b_r  +  x·W_slᵀ + b_sl
// One wave per 16x16 output tile, V_WMMA_F32_16X16X4_F32 over K=128 for the
// 4 relations plus the self-loop (treated as relation #4 with A = x).
// 100000/16 = 6250 M-tiles, 128/16 = 8 N-tiles -> 50000 tiles, 8 waves/block.
// No divergent control flow: EXEC is all-ones as WMMA requires.
// ---------------------------------------------------------------------------
__global__ void rgcn_wmma_gemm_kernel(const float* __restrict__ x,
                                      const float* __restrict__ W,    // [NREL][DIM][DIM]
                                      const float* __restrict__ b,    // [NREL][DIM]
                                      const float* __restrict__ W_sl, // [DIM][DIM]
                                      const float* __restrict__ b_sl, // [DIM]
                                      const float* __restrict__ h,    // [NREL][NNODES][DIM]
                                      const float* __restrict__ deg,  // [NREL][NNODES]
                                      float* __restrict__ out) {      // [NNODES][DIM]
  const int wave_in_block = threadIdx.x >> 5;
  const int lane          = threadIdx.x & 31;
  const int tile          = blockIdx.x * 8 + wave_in_block; // 0..49999
  const int m_tile        = tile >> 3;
  const int n_tile        = tile & 7;
  const int node_base     = m_tile * 16;
  const int feat_base     = n_tile * 16;
  const int half          = lane >> 4;   // selects K pair (A/B) and M offset (C/D)
  const int mn            = lane & 15;   // A row / B column within tile

  v8f c = {0.f, 0.f, 0.f, 0.f, 0.f, 0.f, 0.f, 0.f};

  const float* Abase[5];
  const float* Wbase[5];
#pragma unroll
  for (int r = 0; r < NREL; ++r) {
    Abase[r] = h + (size_t)r * NNODES * DIM;
    Wbase[r] = W + (size_t)r * DIM * DIM;
  }
  Abase[4] = x;
  Wbase[4] = W_sl;

#pragma unroll
  for (int r = 0; r < 5; ++r) {
    // A fragment: lane (mn, half) holds {K=k+2*half, K=k+2*half+1} of row mn.
    const float* Aptr = Abase[r] + (size_t)(node_base + mn) * DIM + 2 * half;
    // B fragment: symmetric layout over columns of W[f][k] (row-major f,k).
    const float* Bptr = Wbase[r] + (size_t)(feat_base + mn) * DIM + 2 * half;
#pragma unroll 8
    for (int k = 0; k < DIM; k += 4) {
      v2f a  = *reinterpret_cast<const v2f*>(Aptr + k);
      v2f bb = *reinterpret_cast<const v2f*>(Bptr + k);
      c = __builtin_amdgcn_wmma_f32_16x16x4_f32(
          /*neg_a=*/false, a, /*neg_b=*/false, bb,
          /*c_mod=*/(short)0, c, /*reuse_a=*/false, /*reuse_b=*/false);
    }
  }

  // Bias terms + store. C/D layout: VGPR v -> M = v + 8*half, N = mn.
  const int feat = feat_base + mn;
  const float bsl = b_sl[feat];
  const float br0 = b[0 * DIM + feat];
  const float br1 = b[1 * DIM + feat];
  const float br2 = b[2 * DIM + feat];
  const float br3 = b[3 * DIM + feat];
#pragma unroll
  for (int v = 0; v < 8; ++v) {
    const int row = node_base + v + 8 * half;
    float add = bsl
              + deg[0 * NNODES + row] * br0
              + deg[1 * NNODES + row] * br1
              + deg[2 * NNODES + row] * br2
              + deg[3 * NNODES + row] * br3;
    out[(size_t)row * DIM + feat] = c[v] + add;
  }
}

extern "C" void kernel_launch(void* const* d_in, const int* in_sizes, int n_in,
                              void* d_out, int out_size, void* d_ws, size_t ws_size,
                              hipStream_t stream) {
  (void)in_sizes; (void)n_in; (void)out_size; (void)ws_size;

  const float* x     = (const float*)d_in[0];
  const int*   edges = (const int*)d_in[1];
  const float* W     = (const float*)d_in[2];
  const float* b     = (const float*)d_in[3];
  const float* W_sl  = (const float*)d_in[4];
  const float* b_sl  = (const float*)d_in[5];
  float*       out   = (float*)d_out;

  // Workspace layout: h[NREL][NNODES][DIM] f32, then deg[NREL][NNODES] f32.
  float* h   = (float*)d_ws;
  float* deg = h + (size_t)NREL * NNODES * DIM;
  const size_t zero_bytes =
      ((size_t)NREL * NNODES * DIM + (size_t)NREL * NNODES) * sizeof(float);
  hipMemsetAsync(d_ws, 0, zero_bytes, stream);  // graph-capturable memset node

  // Scatter: NEDGES waves per relation = 500000*32 threads = 62500 blocks of 256.
  dim3 sgrid(62500, NREL);
  rgcn_scatter_kernel<<<sgrid, 256, 0, stream>>>(x, edges, h, deg);

  // GEMM: 50000 tiles / 8 waves per block = 6250 blocks (exact).
  rgcn_wmma_gemm_kernel<<<6250, 256, 0, stream>>>(x, W, b, W_sl, b_sl, h, deg, out);
}